// simCLR_66872640798944
// MI455X (gfx1250) — compile-verified
//
#include <hip/hip_runtime.h>

// SimCLR NT-Xent loss, N=4096, D=128, T=0.1 — f32 WMMA (v_wmma_f32_16x16x4_f32)
// sim matrix (8192x8192) is never materialized: fused exp + row-sum.

typedef __attribute__((ext_vector_type(2))) float v2f;
typedef __attribute__((ext_vector_type(8))) float v8f;

#define N_ROWS 4096
#define DIM    128
#define TWO_N  8192
#define INV_T  10.0f

__device__ __forceinline__ float wave_reduce_sum(float v) {
    #pragma unroll
    for (int off = 16; off > 0; off >>= 1)
        v += __shfl_xor(v, off, 32);
    return v;
}

// ---------------------------------------------------------------------------
// Kernel 1: L2-normalize anchor & target rows into z[0:N] / z[N:2N], and
// compute positives[r] = dot(a_hat_r, t_hat_r).  One wave (32 lanes) per row:
// each lane owns 4 of the 128 columns (one float4).
// ---------------------------------------------------------------------------
__global__ __launch_bounds__(256) void norm_pos_kernel(
    const float* __restrict__ A, const float* __restrict__ T,
    float* __restrict__ z, float* __restrict__ pos) {
    const int wave = threadIdx.x >> 5;
    const int lane = threadIdx.x & 31;
    const int r = blockIdx.x * 8 + wave;   // grid sized exactly: r < N_ROWS

    const float4 av = ((const float4*)(A + (size_t)r * DIM))[lane];
    float sa = av.x * av.x + av.y * av.y + av.z * av.z + av.w * av.w;
    sa = wave_reduce_sum(sa);
    const float ra = rsqrtf(fmaxf(sa, 1e-12f));
    const float4 an = make_float4(av.x * ra, av.y * ra, av.z * ra, av.w * ra);
    ((float4*)(z + (size_t)r * DIM))[lane] = an;

    const float4 tv = ((const float4*)(T + (size_t)r * DIM))[lane];
    float st = tv.x * tv.x + tv.y * tv.y + tv.z * tv.z + tv.w * tv.w;
    st = wave_reduce_sum(st);
    const float rt = rsqrtf(fmaxf(st, 1e-12f));
    const float4 tn = make_float4(tv.x * rt, tv.y * rt, tv.z * rt, tv.w * rt);
    ((float4*)(z + (size_t)(r + N_ROWS) * DIM))[lane] = tn;

    float d = an.x * tn.x + an.y * tn.y + an.z * tn.z + an.w * tn.w;
    d = wave_reduce_sum(d);
    if (lane == 0) pos[r] = d;
}

// ---------------------------------------------------------------------------
// Kernel 2: denom[i] = sum_{j != i} exp(10 * z_i . z_j).
// One workgroup per 16-row strip of sim; 8 waves split the 512 j-tiles.
// Each 16x16 tile = 32 chained V_WMMA_F32_16X16X4_F32 over K=128.
// A/B operand layout (ISA 7.12.2, 32-bit 16x4 / 4x16): lane L holds row
// base+(L&15), k-cols 2*(L>>4)+{0,1} per VGPR pair -> one float2 load each.
// C/D layout: VGPR v, half h -> row v+8h, col L&15.
// ---------------------------------------------------------------------------
__global__ __launch_bounds__(256) void denom_kernel(
    const float* __restrict__ z, float* __restrict__ denom) {
    __shared__ float s_denom[16];

    const int lane = threadIdx.x & 31;
    const int wave = threadIdx.x >> 5;
    const int i0 = blockIdx.x * 16;

    if (threadIdx.x < 16) s_denom[threadIdx.x] = 0.0f;
    __syncthreads();

    const int half = lane >> 4;
    const int l16  = lane & 15;
    const int colb = half * 2;               // k-offset within each 4-wide step

    // Preload this strip's A operand slices for all 32 K-steps (64 VGPRs).
    v2f Areg[32];
    const float* abase = z + (size_t)(i0 + l16) * DIM + colb;
    #pragma unroll
    for (int k = 0; k < 32; ++k)
        Areg[k] = *(const v2f*)(abase + k * 4);

    v8f acc = {};                            // per-lane exp-sum accumulators
    for (int jt = wave; jt < TWO_N / 16; jt += 8) {   // uniform per wave
        const int j0 = jt * 16;
        const float* bbase = z + (size_t)(j0 + l16) * DIM + colb;
        v8f c = {};
        #pragma unroll
        for (int k = 0; k < 32; ++k) {
            v2f b = *(const v2f*)(bbase + k * 4);
            c = __builtin_amdgcn_wmma_f32_16x16x4_f32(
                    false, Areg[k], false, b, (short)0, c, false, false);
        }
        const int gj = j0 + l16;             // this lane's global column
        #pragma unroll
        for (int v = 0; v < 8; ++v) {
            const int gi = i0 + v + 8 * half;        // global row of c[v]
            const float e = (gi == gj) ? 0.0f : __expf(c[v] * INV_T);
            acc[v] += e;
        }
    }

    // Row sums: columns of one row live across a 16-lane half.
    #pragma unroll
    for (int v = 0; v < 8; ++v) {
        float s = acc[v];
        #pragma unroll
        for (int off = 1; off < 16; off <<= 1)
            s += __shfl_xor(s, off, 32);     // stays within each half
        acc[v] = s;
    }
    if (l16 == 0) {                          // lane 0: rows 0..7, lane 16: rows 8..15
        #pragma unroll
        for (int v = 0; v < 8; ++v)
            atomicAdd(&s_denom[v + 8 * half], acc[v]);
    }
    __syncthreads();
    if (threadIdx.x < 16)
        denom[i0 + threadIdx.x] = s_denom[threadIdx.x];
}

// ---------------------------------------------------------------------------
// Kernel 3: loss = mean_i( log(denom[i]) - positives[i]/T ), single block.
// ---------------------------------------------------------------------------
__global__ __launch_bounds__(256) void loss_kernel(
    const float* __restrict__ denom, const float* __restrict__ pos,
    float* __restrict__ out) {
    __shared__ float red[256];
    float local = 0.0f;
    for (int i = threadIdx.x; i < TWO_N; i += 256) {
        const float p = pos[i & (N_ROWS - 1)];   // positives repeat for both halves
        local += __logf(denom[i]) - p * INV_T;
    }
    red[threadIdx.x] = local;
    __syncthreads();
    #pragma unroll
    for (int s = 128; s > 0; s >>= 1) {
        if (threadIdx.x < s) red[threadIdx.x] += red[threadIdx.x + s];
        __syncthreads();
    }
    if (threadIdx.x == 0) out[0] = red[0] / (float)TWO_N;
}

// ---------------------------------------------------------------------------
extern "C" void kernel_launch(void* const* d_in, const int* in_sizes, int n_in,
                              void* d_out, int out_size, void* d_ws, size_t ws_size,
                              hipStream_t stream) {
    (void)in_sizes; (void)n_in; (void)out_size; (void)ws_size;
    const float* A = (const float*)d_in[0];   // anchor_embeddings [4096,128] f32
    const float* T = (const float*)d_in[1];   // target_embeddings [4096,128] f32

    float* ws    = (float*)d_ws;
    float* z     = ws;                         // [8192*128] normalized concat
    float* pos   = z + (size_t)TWO_N * DIM;    // [4096]
    float* denom = pos + N_ROWS;               // [8192]

    norm_pos_kernel<<<N_ROWS / 8, 256, 0, stream>>>(A, T, z, pos);
    denom_kernel<<<TWO_N / 16, 256, 0, stream>>>(z, denom);
    loss_kernel<<<1, 256, 0, stream>>>(denom, pos, (float*)d_out);
}